// SparseMixtureOfExperts_16535624090068
// MI455X (gfx1250) — compile-verified
//
#include <hip/hip_runtime.h>
#include <hip/hip_bf16.h>

#define N_TOK 4096
#define DIM   2048
#define ODIM  2048
#define NEXP  8

#define BM 128
#define BN 128
#define BK 32
#define ASTR 36    // padded LDS row stride (dwords) for A: conflict-free wmma reads
#define BSTR 136   // padded LDS row stride (dwords) for B: conflict-free wmma reads

typedef __attribute__((ext_vector_type(2))) float v2f;
typedef __attribute__((ext_vector_type(8))) float v8f;

static __device__ __forceinline__ void atomAddF(float* p, float v) {
  __hip_atomic_fetch_add(p, v, __ATOMIC_RELAXED, __HIP_MEMORY_SCOPE_AGENT);
}

// ---------------------------------------------------------------- zero out+counters
__global__ void moe_zero(float* __restrict__ out, int* __restrict__ counts) {
  size_t i = (size_t)blockIdx.x * blockDim.x + threadIdx.x;   // 2M float4 exactly
  float4 z; z.x = 0.f; z.y = 0.f; z.z = 0.f; z.w = 0.f;
  ((float4*)out)[i] = z;
  if (blockIdx.x == 0 && threadIdx.x < NEXP) counts[threadIdx.x] = 0;
}

// ---------------------------------------------------------------- routing (top-2)
__global__ __launch_bounds__(256) void moe_route(
    const float* __restrict__ x, const float* __restrict__ Wg,
    const float* __restrict__ bg,
    int* __restrict__ counts, int* __restrict__ tokL, float* __restrict__ wtL) {
  __shared__ float wg_s[1024 * NEXP];               // 32 KB half of Wg
  const int lane = threadIdx.x & 31;
  const int wave = threadIdx.x >> 5;
  const int tok  = blockIdx.x * 8 + wave;           // one wave32 per token

  float acc[NEXP];
#pragma unroll
  for (int e = 0; e < NEXP; ++e) acc[e] = 0.f;

  for (int dbase = 0; dbase < DIM; dbase += 1024) {
    __syncthreads();
    const float4* src = (const float4*)(Wg + (size_t)dbase * NEXP);
    float4* dst = (float4*)wg_s;
    for (int i = threadIdx.x; i < 1024 * NEXP / 4; i += 256) dst[i] = src[i];
    __syncthreads();
    for (int i = lane; i < 1024; i += 32) {
      float xv = x[(size_t)tok * DIM + dbase + i];
      const float* w = &wg_s[i * NEXP];
#pragma unroll
      for (int e = 0; e < NEXP; ++e) acc[e] += xv * w[e];
    }
  }
  // butterfly reduce across the wave (warpSize==32 on gfx1250)
#pragma unroll
  for (int e = 0; e < NEXP; ++e) {
    float v = acc[e];
    for (int off = 16; off > 0; off >>= 1) v += __shfl_xor(v, off, 32);
    acc[e] = v;
  }
  if (lane == 0) {
#pragma unroll
    for (int e = 0; e < NEXP; ++e) acc[e] += bg[e];
    int i0 = 0; float v0 = acc[0];
#pragma unroll
    for (int e = 1; e < NEXP; ++e) if (acc[e] > v0) { v0 = acc[e]; i0 = e; }
    int i1 = -1; float v1 = -__builtin_inff();
#pragma unroll
    for (int e = 0; e < NEXP; ++e) {
      if (e == i0) continue;
      if (acc[e] > v1) { v1 = acc[e]; i1 = e; }
    }
    // softmax over {v0, v1}, v0 >= v1  (stable)
    float ex = expf(v1 - v0);
    float w1 = ex / (1.f + ex);
    float w0 = 1.f - w1;
    int s0 = atomicAdd(&counts[i0], 1);
    tokL[i0 * N_TOK + s0] = tok;  wtL[i0 * N_TOK + s0] = w0;
    int s1 = atomicAdd(&counts[i1], 1);
    tokL[i1 * N_TOK + s1] = tok;  wtL[i1 * N_TOK + s1] = w1;
  }
}

// ---------------------------------------------------------------- expert GEMM (fp32 WMMA)
__global__ __launch_bounds__(256) void moe_gemm(
    const float* __restrict__ x, const float* __restrict__ We,
    const float* __restrict__ be,
    const int* __restrict__ counts, const int* __restrict__ tokL,
    const float* __restrict__ wtL, float* __restrict__ out) {
  const int e   = blockIdx.z;
  const int cnt = counts[e];
  const int rb  = blockIdx.y * BM;
  if (rb >= cnt) return;                       // block-uniform exit
  const int c0  = blockIdx.x * BN;

  __shared__ __align__(16) float As[BM * ASTR];   // 18.0 KB
  __shared__ __align__(16) float Bs[BK * BSTR];   // 17.0 KB

  const int*   tokE = tokL + e * N_TOK;
  const float* wtE  = wtL + e * N_TOK;
  const float* B    = We + (size_t)e * DIM * ODIM;

  const int tid  = threadIdx.x;
  const int lane = tid & 31, wave = tid >> 5;
  const int wm = wave & 3, wn = wave >> 2;     // 4x2 wave grid: 32 rows x 64 cols each
  const int h  = lane >> 4, n16 = lane & 15;

  // A-tile loader assignment: 2 threads per row, 16 contiguous floats each
  const int arow = tid >> 1;
  const int acol = (tid & 1) * 16;
  const int gra  = rb + arow;
  const int tokA = (gra < cnt) ? tokE[gra] : 0;
  const float* xrow = x + (size_t)tokA * DIM + acol;

  v8f acc[2][4];
#pragma unroll
  for (int mt = 0; mt < 2; ++mt)
#pragma unroll
    for (int nt = 0; nt < 4; ++nt)
#pragma unroll
      for (int r = 0; r < 8; ++r) acc[mt][nt][r] = 0.f;

  for (int kb = 0; kb < DIM; kb += BK) {
    // stage global loads in registers before the barrier
    float4 av[4], bv[4];
#pragma unroll
    for (int i = 0; i < 4; ++i)
      av[i] = *(const float4*)(xrow + kb + i * 4);
#pragma unroll
    for (int i = 0; i < 4; ++i) {
      int q = tid + i * 256;                   // 1024 float4 of B tile
      int r = q >> 5, c4 = (q & 31) * 4;
      bv[i] = *(const float4*)(B + (size_t)(kb + r) * ODIM + c0 + c4);
    }
    __syncthreads();                           // previous compute done
#pragma unroll
    for (int i = 0; i < 4; ++i)
      *(float4*)&As[arow * ASTR + acol + i * 4] = av[i];
#pragma unroll
    for (int i = 0; i < 4; ++i) {
      int q = tid + i * 256;
      int r = q >> 5, c4 = (q & 31) * 4;
      *(float4*)&Bs[r * BSTR + c4] = bv[i];
    }
    __syncthreads();

#pragma unroll
    for (int ks = 0; ks < BK / 4; ++ks) {
      const int k0 = ks * 4;
      // A fragment (16x4 f32): lane h half selects K pair {2h, 2h+1}
      v2f a0 = *(const v2f*)&As[(wm * 32 +      n16) * ASTR + k0 + 2 * h];
      v2f a1 = *(const v2f*)&As[(wm * 32 + 16 + n16) * ASTR + k0 + 2 * h];
#pragma unroll
      for (int nt = 0; nt < 4; ++nt) {
        const int c = wn * 64 + nt * 16 + n16;
        v2f b;
        b.x = Bs[(k0 + 2 * h)     * BSTR + c];
        b.y = Bs[(k0 + 2 * h + 1) * BSTR + c];
        acc[0][nt] = __builtin_amdgcn_wmma_f32_16x16x4_f32(
            false, a0, false, b, (short)0, acc[0][nt], false, false);
        acc[1][nt] = __builtin_amdgcn_wmma_f32_16x16x4_f32(
            false, a1, false, b, (short)0, acc[1][nt], false, false);
      }
    }
  }

  // epilogue: + bias, * routing weight, scatter-accumulate (2 adds/elem total -> deterministic)
  const int rowbase = rb + wm * 32;
#pragma unroll
  for (int mt = 0; mt < 2; ++mt) {
    int   tokid[8]; float wgt[8]; bool ok[8];
#pragma unroll
    for (int r = 0; r < 8; ++r) {
      int g = rowbase + mt * 16 + 8 * h + r;   // C layout: VGPR r -> row r + 8h
      ok[r]    = (g < cnt);
      tokid[r] = ok[r] ? tokE[g] : 0;
      wgt[r]   = ok[r] ? wtE[g] : 0.f;
    }
#pragma unroll
    for (int nt = 0; nt < 4; ++nt) {
      const int col  = c0 + wn * 64 + nt * 16 + n16;
      const float bi = be[e * ODIM + col];
#pragma unroll
      for (int r = 0; r < 8; ++r) {
        if (ok[r])
          atomAddF(&out[(size_t)tokid[r] * ODIM + col], wgt[r] * (acc[mt][nt][r] + bi));
      }
    }
  }
}

// ---------------------------------------------------------------- launch
extern "C" void kernel_launch(void* const* d_in, const int* in_sizes, int n_in,
                              void* d_out, int out_size, void* d_ws, size_t ws_size,
                              hipStream_t stream) {
  const float* x  = (const float*)d_in[0];
  const float* Wg = (const float*)d_in[1];
  const float* bg = (const float*)d_in[2];
  const float* We = (const float*)d_in[3];
  const float* be = (const float*)d_in[4];
  float* out = (float*)d_out;

  char* ws = (char*)d_ws;
  int*   counts = (int*)ws;                                        // 8 ints
  int*   tokL   = (int*)(ws + 32);                                 // 8*4096 ints
  float* wtL    = (float*)(ws + 32 + NEXP * N_TOK * sizeof(int));  // 8*4096 floats

  const int zero_blocks = (int)(((size_t)N_TOK * ODIM / 4) / 256); // exact
  moe_zero<<<zero_blocks, 256, 0, stream>>>(out, counts);
  moe_route<<<N_TOK / 8, 256, 0, stream>>>(x, Wg, bg, counts, tokL, wtL);
  dim3 grid(ODIM / BN, (N_TOK + BM - 1) / BM, NEXP);
  moe_gemm<<<grid, 256, 0, stream>>>(x, We, be, counts, tokL, wtL, out);
}